// Low_Rank_RNN_39290360824511
// MI455X (gfx1250) — compile-verified
//
#include <hip/hip_runtime.h>
#include <hip/hip_bf16.h>
#include <cmath>

typedef __attribute__((ext_vector_type(2))) float v2f;
typedef __attribute__((ext_vector_type(8))) float v8f;

#define B_ 64
#define T_ 512
#define I_ 16
#define H_ 1024
#define O_ 8
#define R_ 4
#define NOISE_STD_ 0.05f
#define TAU_ 0.2f

// ---------------------------------------------------------------------------
// Kernel 1: inp[(b*T+t)*H + h] = sum_i u[(b*T+t)*I + i] * Win_w[h*I + i] + Win_b[h]
// GEMM: M = B*T = 32768, N = H = 1024, K = I = 16.
// One wave computes a 16x16 f32 tile via 4 chained V_WMMA_F32_16X16X4_F32.
//
// f32 WMMA operand layouts (ISA 7.12.2):
//   A 16x4 : lane L<16 -> row M=L, VGPR0=K0,VGPR1=K1 ; lane L>=16 -> M=L-16, K2/K3
//   B 4x16 : lane L<16 -> col N=L, VGPR0=K0,VGPR1=K1 ; lane L>=16 -> N=L-16, K2/K3
//   C 16x16: VGPR v, lane L<16 -> M=v,  N=L ; lane L>=16 -> M=v+8, N=L-16
// ---------------------------------------------------------------------------
__global__ void __launch_bounds__(256) win_gemm_wmma(
    const float* __restrict__ u, const float* __restrict__ Win_w,
    const float* __restrict__ Win_b, float* __restrict__ inp)
{
  const int lane  = threadIdx.x & 31;
  const int wave  = threadIdx.x >> 5;
  const int tile  = blockIdx.x * 8 + wave;     // 131072 tiles total
  const int rowTile = tile >> 6;               // 2048 tiles over B*T
  const int colTile = tile & 63;               // 64 tiles over H

  const int r16   = lane & 15;                 // M for A, N for B/C
  const int khalf = lane >> 4;                 // selects K pair

  const int gRowBase = rowTile * 16;           // row index into flattened (B*T)
  const int hBase    = colTile * 16;

  v8f c = {};
#pragma unroll
  for (int kt = 0; kt < 4; ++kt) {
    const int k = kt * 4 + khalf * 2;
    v2f a, bf;
    // A fragment from u (row-major, I=16 contiguous)
    const float* ap = u + (size_t)(gRowBase + r16) * I_ + k;
    a.x = ap[0]; a.y = ap[1];
    // B fragment: B[k][n] = Win_w[hBase+n][k]  (Win_w is (H,I) row-major)
    const float* bp = Win_w + (size_t)(hBase + r16) * I_ + k;
    bf.x = bp[0]; bf.y = bp[1];
    c = __builtin_amdgcn_wmma_f32_16x16x4_f32(false, a, false, bf,
                                              (short)0, c, false, false);
  }

  const float bias = Win_b[hBase + r16];
#pragma unroll
  for (int v = 0; v < 8; ++v) {
    const int m = v + khalf * 8;
    inp[(size_t)(gRowBase + m) * H_ + hBase + r16] = c[v] + bias;
  }
}

// ---------------------------------------------------------------------------
// Combined 12-value block reduction (1024 threads = 32 waves, wave32).
// vals[0..3]  -> z = r @ N            (needed by every thread next step)
// vals[4..11] -> y = r @ Wout_w.T     (written by threads 0..7)
// Result broadcast through sm_res[12]. Two barriers per call.
// ---------------------------------------------------------------------------
__device__ __forceinline__ void block_reduce12(float vals[12],
                                               float* __restrict__ sm_part,
                                               float* __restrict__ sm_res,
                                               int lane, int wave, int tid)
{
#pragma unroll
  for (int v = 0; v < 12; ++v) {
    float s = vals[v];
#pragma unroll
    for (int off = 16; off >= 1; off >>= 1) s += __shfl_xor(s, off, 32);
    vals[v] = s;
  }
  if (lane == 0) {
#pragma unroll
    for (int v = 0; v < 12; ++v) sm_part[wave * 12 + v] = vals[v];
  }
  __syncthreads();
  if (tid < 12) {
    float s = 0.0f;
#pragma unroll
    for (int w = 0; w < 32; ++w) s += sm_part[w * 12 + tid];
    sm_res[tid] = s;
  }
  __syncthreads();
}

// ---------------------------------------------------------------------------
// Kernel 2: fused scan + output projection. One workgroup per batch element;
// thread h owns state x[h] in a register. Per step:
//   z_t (from previous step's reduction) -> rec -> x update -> r = tanh(x)
//   -> accumulate z_{t+1} and y_t partials -> one combined reduction.
// y_t uses the SAME tanh as the next step's recurrence (traj[t] == x_{t+1}),
// so traj (256 MB of traffic) never touches HBM.
// noise is strictly streaming (zero reuse) -> non-temporal loads keep L2 free
// for the inp producer->consumer window.
// ---------------------------------------------------------------------------
__global__ void __launch_bounds__(1024) rnn_scan_fused(
    const float* __restrict__ x0, const float* __restrict__ noise,
    const float* __restrict__ M, const float* __restrict__ N,
    const float* __restrict__ Wout_w, const float* __restrict__ Wout_b,
    const float* __restrict__ inp, float* __restrict__ out)
{
  const int b    = blockIdx.x;
  const int h    = threadIdx.x;
  const int lane = h & 31;
  const int wave = h >> 5;

  __shared__ float sm_part[32 * 12];
  __shared__ float sm_res[12];

  float Mr[R_], Nr[R_];
#pragma unroll
  for (int j = 0; j < R_; ++j) { Mr[j] = M[h * R_ + j]; Nr[j] = N[h * R_ + j]; }
  float Wr[O_];
#pragma unroll
  for (int o = 0; o < O_; ++o) Wr[o] = Wout_w[o * H_ + h];
  const float wb = (h < O_) ? Wout_b[h] : 0.0f;   // hoisted out of the T loop

  float x = x0[b * H_ + h];
  const float scale = 1.0f / (float)H_;

  // Partials of r_0 = tanh(x0): z-part valid, y-part unused (t = -1 never written)
  float vals[12];
  {
    float r = tanhf(x);
#pragma unroll
    for (int j = 0; j < R_; ++j) vals[j] = r * Nr[j];
#pragma unroll
    for (int o = 0; o < O_; ++o) vals[4 + o] = 0.0f;
  }

  const float* inp_b = inp + (size_t)b * T_ * H_;

  // Prefetch t = 0 operands (noise: non-temporal streaming)
  float inp_v = inp_b[h];
  float nz    = __builtin_nontemporal_load(&noise[(size_t)b * H_ + h]);

  for (int t = 0; t < T_; ++t) {
    // Reduction of r_t partials -> z_t (all threads) and y_{t-1} (threads 0..7)
    block_reduce12(vals, sm_part, sm_res, lane, wave, h);
    if (t > 0 && h < O_)
      out[((size_t)b * T_ + (t - 1)) * O_ + h] = sm_res[4 + h] + wb;
    const float z0 = sm_res[0], z1 = sm_res[1], z2 = sm_res[2], z3 = sm_res[3];

    // Consume prefetched operands for step t
    const float cur_inp = inp_v;
    const float cur_nz  = nz;
    if (t + 1 < T_) {   // issue next step's loads; latency hides under tanh+reduce
      inp_v = inp_b[(size_t)(t + 1) * H_ + h];
      nz    = __builtin_nontemporal_load(
                  &noise[((size_t)(t + 1) * B_ + b) * H_ + h]);
    }

    // x_{t+1} = x_t + noise_std*n + tau*(-x_t + (z @ M.T)/H + inp_t)
    const float rec = scale * (z0 * Mr[0] + z1 * Mr[1] + z2 * Mr[2] + z3 * Mr[3])
                    + cur_inp;
    x = x + NOISE_STD_ * cur_nz + TAU_ * (rec - x);

    // r_{t+1}: feeds both z_{t+1} (recurrence) and y_t (output projection)
    const float r = tanhf(x);
#pragma unroll
    for (int j = 0; j < R_; ++j) vals[j] = r * Nr[j];
#pragma unroll
    for (int o = 0; o < O_; ++o) vals[4 + o] = r * Wr[o];
  }

  // Final reduction -> y_{T-1}
  block_reduce12(vals, sm_part, sm_res, lane, wave, h);
  if (h < O_)
    out[((size_t)b * T_ + (T_ - 1)) * O_ + h] = sm_res[4 + h] + wb;
}

// ---------------------------------------------------------------------------
extern "C" void kernel_launch(void* const* d_in, const int* in_sizes, int n_in,
                              void* d_out, int out_size, void* d_ws, size_t ws_size,
                              hipStream_t stream)
{
  const float* u      = (const float*)d_in[0];  // (B,T,I)
  const float* x0     = (const float*)d_in[1];  // (B,H)
  const float* noise  = (const float*)d_in[2];  // (T,B,H)
  const float* M      = (const float*)d_in[3];  // (H,R)
  const float* N      = (const float*)d_in[4];  // (H,R)
  const float* Win_w  = (const float*)d_in[5];  // (H,I)
  const float* Win_b  = (const float*)d_in[6];  // (H)
  const float* Wout_w = (const float*)d_in[7];  // (O,H)
  const float* Wout_b = (const float*)d_in[8];  // (O)
  float* out = (float*)d_out;                   // (B,T,O) f32
  float* inp = (float*)d_ws;                    // (B*T,H) f32 = 128 MB scratch

  // Input GEMM on WMMA: 2048 x 64 tiles, 8 waves per 256-thread block
  win_gemm_wmma<<<(2048 * 64) / 8, 256, 0, stream>>>(u, Win_w, Win_b, inp);

  // Persistent fused scan: one workgroup per batch element
  rnn_scan_fused<<<B_, 1024, 0, stream>>>(x0, noise, M, N, Wout_w, Wout_b,
                                          inp, out);
}